// DetectionLoss_50989851738817
// MI455X (gfx1250) — compile-verified
//
#include <hip/hip_runtime.h>
#include <hip/hip_bf16.h>
#include <stdint.h>

#define NCLS       80
#define CSTRIDE    85          // 5 + NUM_CLASSES floats per pred row
#define TTGT       64
#define LBOX       5.0f
#define SPP        512         // preds staged per block
#define THREADS    256

typedef __attribute__((ext_vector_type(2))) float v2f;
typedef __attribute__((ext_vector_type(8))) float v8f;

// ---------------------------------------------------------------- init ------
__global__ void dl_init(unsigned long long* __restrict__ keys,
                        float* __restrict__ out, int nkeys) {
  int i = blockIdx.x * blockDim.x + threadIdx.x;
  if (i == 0) out[0] = 0.0f;
  for (int k = i; k < nkeys; k += gridDim.x * blockDim.x) keys[k] = 0ull;
}

// ------------------------------------------------- CDNA5 async copy helpers -
__device__ __forceinline__ void async_g2l_b128(uint32_t lds_byte_off,
                                               const void* gaddr) {
  // gfx1250 ASYNCcnt-tracked DMA: global memory -> LDS, no VGPR staging.
  asm volatile("global_load_async_to_lds_b128 %0, %1, off"
               :: "v"(lds_byte_off), "v"(gaddr) : "memory");
}
__device__ __forceinline__ void wait_asynccnt0() {
  asm volatile("s_wait_asynccnt 0" ::: "memory");
}

__device__ __forceinline__ float softplusf(float x) {
  return fmaxf(x, 0.0f) + log1pf(expf(-fabsf(x)));
}

// -------------------------------------- wave32 sum via the matrix pipe ------
// V_WMMA_F32_16X16X4_F32 with B == ones, C == 0:
//   D[m,n] = sum_k A[m,k]  =>  every column of D sums to total(A).
// Each lane contributes its partial in a.x (a.y = 0). A lane's 8 D VGPRs are
// half of one column; shfl_xor(16) folds the two halves -> exact f32 wave sum
// in every lane. Requires EXEC all-ones (call only from converged code).
__device__ __forceinline__ float wave_sum_wmma(float x) {
  v2f a; a.x = x;    a.y = 0.0f;
  v2f b; b.x = 1.0f; b.y = 1.0f;
  v8f c = {};
  v8f d = __builtin_amdgcn_wmma_f32_16x16x4_f32(
      /*neg_a=*/false, a, /*neg_b=*/false, b,
      /*c_mod=*/(short)0, c, /*reuse_a=*/false, /*reuse_b=*/false);
  float s = ((d[0] + d[1]) + (d[2] + d[3])) + ((d[4] + d[5]) + (d[6] + d[7]));
  s += __shfl_xor(s, 16, 32);
  return s;
}

// ------------------------------------------------------ main match kernel ---
__global__ __launch_bounds__(THREADS)
void dl_match(const float* __restrict__ raw, const float* __restrict__ tgt,
              unsigned long long* __restrict__ keys, float* __restrict__ out,
              int P, float invB) {
  __shared__ __align__(16) float s_box[SPP * 4];   // staged pred boxes
  __shared__ float s_tb[TTGT * 4];                 // target boxes
  __shared__ float s_ta[TTGT];                     // target areas
  __shared__ unsigned long long s_key[TTGT];       // per-target best key
  __shared__ float s_red[THREADS / 32];

  const int b    = blockIdx.y;
  const int tid  = threadIdx.x;
  const int p0   = blockIdx.x * SPP;
  const float* rawb = raw + (size_t)b * P * CSTRIDE;

  // targets -> LDS (tiny: 64 rows of 5 floats)
  if (tid < TTGT) {
    const float* tr = tgt + ((size_t)b * TTGT + tid) * 5;
    float x1 = tr[0], y1 = tr[1], x2 = tr[2], y2 = tr[3];
    s_tb[tid * 4 + 0] = x1; s_tb[tid * 4 + 1] = y1;
    s_tb[tid * 4 + 2] = x2; s_tb[tid * 4 + 3] = y2;
    s_ta[tid] = (x2 - x1) * (y2 - y1);
    s_key[tid] = 0ull;
  }

  // stage pred boxes via async DMA; fold softplus(conf) on the side
  float conf_sum = 0.0f;
  for (int k = tid; k < SPP; k += THREADS) {
    int p = p0 + k;
    if (p < P) {
      const float* pr = rawb + (size_t)p * CSTRIDE;
      async_g2l_b128((uint32_t)(uintptr_t)&s_box[k * 4], pr);  // box: 4 floats
      conf_sum += softplusf(pr[4]);                            // conf logit
    } else {
      s_box[k * 4 + 0] = 1e30f; s_box[k * 4 + 1] = 1e30f;      // degenerate
      s_box[k * 4 + 2] = -1e30f; s_box[k * 4 + 3] = -1e30f;
    }
  }
  wait_asynccnt0();
  __syncthreads();

  // each thread owns target t, scans its quarter of the staged preds
  const int t   = tid & (TTGT - 1);
  const int sub = tid >> 6;                         // 0..3
  const float tx1 = s_tb[t * 4 + 0], ty1 = s_tb[t * 4 + 1];
  const float tx2 = s_tb[t * 4 + 2], ty2 = s_tb[t * 4 + 3];
  const float ta  = s_ta[t];

  unsigned long long best = 0ull;
  const int lo = sub * (SPP / 4);
  const int hi = min(lo + (SPP / 4), P - p0);
  for (int k = lo; k < hi; ++k) {
    float px1 = s_box[k * 4 + 0], py1 = s_box[k * 4 + 1];
    float px2 = s_box[k * 4 + 2], py2 = s_box[k * 4 + 3];
    float w = fminf(px2, tx2) - fmaxf(px1, tx1);
    float h = fminf(py2, ty2) - fmaxf(py1, ty1);
    float inter = w * h;
    float pa  = (px2 - px1) * (py2 - py1);
    float uni = pa + ta - inter;
    float iou = (w > 0.0f && h > 0.0f && uni > 0.0f) ? inter / uni : 0.0f;
    // iou >= 0 so float bits order monotonically; ~p makes ties pick lowest p
    unsigned long long key =
        ((unsigned long long)__float_as_uint(iou) << 32) |
        (unsigned int)~(unsigned int)(p0 + k);
    if (key > best) best = key;
  }
  atomicMax(&s_key[t], best);          // ds_max_u64, 4-way per address
  __syncthreads();
  if (tid < TTGT) atomicMax(&keys[b * TTGT + tid], s_key[tid]);

  // block-reduce softplus partials: matrix-pipe wave sum, then LDS combine
  float wsum = wave_sum_wmma(conf_sum);
  if ((tid & 31) == 0) s_red[tid >> 5] = wsum;
  __syncthreads();
  if (tid == 0) {
    float s = 0.0f;
    for (int w = 0; w < THREADS / 32; ++w) s += s_red[w];
    atomicAdd(out, s * invB);
  }
}

// ----------------------------------------------------- gathered loss epilog -
__global__ __launch_bounds__(TTGT)
void dl_final(const float* __restrict__ raw, const float* __restrict__ tgt,
              const unsigned long long* __restrict__ keys,
              float* __restrict__ out, int P, float invB) {
  __shared__ int s_idx[TTGT];
  const int b = blockIdx.x;
  const int t = threadIdx.x;

  unsigned long long key = keys[b * TTGT + t];
  int p = (int)~(unsigned int)(key & 0xffffffffull);
  s_idx[t] = p;
  __syncthreads();

  // one-hot scatter collapses duplicates: subtract conf only for first user
  bool first = true;
  for (int u = 0; u < t; ++u)
    if (s_idx[u] == p) first = false;

  const float* pr = raw + ((size_t)b * P + p) * CSTRIDE;
  const float* tr = tgt + ((size_t)b * TTGT + t) * 5;

  // smooth-L1 over 4 coords
  float box = 0.0f;
  for (int c = 0; c < 4; ++c) {
    float d = fabsf(pr[c] - tr[c]);
    box += (d < 1.0f) ? 0.5f * d * d : d - 0.5f;
  }

  // CE over 80 class logits (log-softmax)
  const float* cl = pr + 5;
  int tc = (int)tr[4];
  float m = -INFINITY;
  for (int c = 0; c < NCLS; ++c) m = fmaxf(m, cl[c]);
  float s = 0.0f;
  for (int c = 0; c < NCLS; ++c) s += expf(cl[c] - m);
  float cls = (m + logf(s)) - cl[tc];

  float term = LBOX * box + cls - (first ? pr[4] : 0.0f);

  // matrix-pipe wave reduction; one atomic per wave (2 per block)
  float wsum = wave_sum_wmma(term * invB);
  if ((t & 31) == 0) atomicAdd(out, wsum);
}

// ---------------------------------------------------------------- launch ----
extern "C" void kernel_launch(void* const* d_in, const int* in_sizes, int n_in,
                              void* d_out, int out_size, void* d_ws, size_t ws_size,
                              hipStream_t stream) {
  const float* raw = (const float*)d_in[0];   // (B,P,85) f32
  const float* tgt = (const float*)d_in[1];   // (B,T,5)  f32  (epoch unused)

  int B = 16, P = 16384;
  if (n_in >= 2 && in_sizes[1] >= TTGT * 5) B = in_sizes[1] / (TTGT * 5);
  if (in_sizes[0] >= B * CSTRIDE)           P = in_sizes[0] / (B * CSTRIDE);
  const float invB = 1.0f / (float)B;

  unsigned long long* keys = (unsigned long long*)d_ws;   // B*T u64 = 8 KB
  float* out = (float*)d_out;

  dl_init<<<1, 256, 0, stream>>>(keys, out, B * TTGT);

  dim3 grid((P + SPP - 1) / SPP, B);
  dl_match<<<grid, THREADS, 0, stream>>>(raw, tgt, keys, out, P, invB);

  dl_final<<<B, TTGT, 0, stream>>>(raw, tgt, keys, out, P, invB);
}